// CQAttention_87222195847963
// MI455X (gfx1250) — compile-verified
//
#include <hip/hip_runtime.h>
#include <hip/hip_bf16.h>

typedef __attribute__((ext_vector_type(2))) float v2f;
typedef __attribute__((ext_vector_type(8))) float v8f;

static constexpr int kB  = 32;
static constexpr int kD  = 128;
static constexpr int kLc = 2048;
static constexpr int kLq = 64;
static constexpr float kNegInf = -1.0e30f;

__device__ inline v8f wmma4(v2f a, v2f b, v8f c) {
  // D = A(16x4 f32) * B(4x16 f32) + C(16x16 f32)
  return __builtin_amdgcn_wmma_f32_16x16x4_f32(false, a, false, b, (short)0, c, false, false);
}

__device__ inline void atomicMaxF(float* addr, float v) {
  if (v >= 0.0f) atomicMax((int*)addr, __float_as_int(v));
  else           atomicMin((unsigned int*)addr, __float_as_uint(v));
}

// Compute the wave's 16x64 strip of scq = (Ct * w3) @ Qt^T with f32 WMMA.
// CtL: [128][129] LDS (rows = local i, cols = d)
// QL : [64][129]  LDS (rows = j, cols = d); already w3-scaled when SCALE=false
template<bool SCALE>
__device__ inline void scq_strip(const float* CtL, const float* QL, const float* wl,
                                 int wv, int l16, int half, v8f Sv[4]) {
  const int arow = (wv * 16 + l16) * 129;
  for (int kb = 0; kb < 32; ++kb) {
    const int k = kb * 4 + half * 2;
    v2f a{};
    a[0] = CtL[arow + k];
    a[1] = CtL[arow + k + 1];
    float w0 = 1.0f, w1 = 1.0f;
    if constexpr (SCALE) { w0 = wl[k]; w1 = wl[k + 1]; }
    for (int jt = 0; jt < 4; ++jt) {
      const int jr = (jt * 16 + l16) * 129;
      v2f bb{};
      if constexpr (SCALE) {
        bb[0] = QL[jr + k] * w0;
        bb[1] = QL[jr + k + 1] * w1;
      } else {
        bb[0] = QL[jr + k];
        bb[1] = QL[jr + k + 1];
      }
      Sv[jt] = wmma4(a, bb, Sv[jt]);
    }
  }
}

// ---------------- init + rank-1 score precompute ----------------

__global__ __launch_bounds__(256) void k_init(float* colmaxW, float* colsumW, float* RW) {
  int t = blockIdx.x * 256 + threadIdx.x;
  if (t < kB * kLq) { colmaxW[t] = -3.0e38f; colsumW[t] = 0.0f; }
  if (t < kB * kLq * kD) RW[t] = 0.0f;
}

__global__ __launch_bounds__(256) void k_sc(const float* __restrict__ C,
                                            const float* __restrict__ w,
                                            float* __restrict__ scW) {
  int t = blockIdx.x * 256 + threadIdx.x;            // t < B*Lc
  int b = t >> 11, i = t & (kLc - 1);
  const float* Cb = C + (size_t)b * kD * kLc;
  float s = 0.0f;
  for (int d = 0; d < kD; ++d) s += Cb[(size_t)d * kLc + i] * w[d];
  scW[t] = s;
}

__global__ __launch_bounds__(64) void k_sq(const float* __restrict__ Q,
                                           const float* __restrict__ w,
                                           float* __restrict__ sqW) {
  int b = blockIdx.x, j = threadIdx.x;
  const float* Qb = Q + (size_t)b * kD * kLq;
  float s = 0.0f;
  for (int d = 0; d < kD; ++d) s += Qb[(size_t)d * kLq + j] * w[kD + d];
  sqW[b * kLq + j] = s;
}

// ---------------- pass 1: row softmax stats + column max ----------------

__global__ __launch_bounds__(256)
void k1_stats(const float* __restrict__ C, const float* __restrict__ Q,
              const float* __restrict__ w, const float* __restrict__ qmask,
              const float* __restrict__ scW, const float* __restrict__ sqW,
              float* __restrict__ rowmaxW, float* __restrict__ rowsumW,
              float* __restrict__ colmaxW) {
  __shared__ float CtL[128 * 129];
  __shared__ float QwL[64 * 129];
  __shared__ float colL[8 * 64];
  const int tid = threadIdx.x;
  const int b = blockIdx.y;
  const int ibase = blockIdx.x * 128;
  const size_t cb = (size_t)b * kD * kLc;
  const size_t qb = (size_t)b * kD * kLq;

  for (int idx = tid; idx < 128 * 128; idx += 256) {
    int i = idx & 127, d = idx >> 7;
    CtL[i * 129 + d] = C[cb + (size_t)d * kLc + ibase + i];
  }
  for (int idx = tid; idx < 64 * 128; idx += 256) {
    int j = idx & 63, d = idx >> 6;
    QwL[j * 129 + d] = Q[qb + (size_t)d * kLq + j] * w[2 * kD + d];
  }
  __syncthreads();

  const int wv = tid >> 5, lane = tid & 31, l16 = lane & 15, half = lane >> 4;
  v8f Sv[4] = {};
  scq_strip<false>(CtL, QwL, nullptr, wv, l16, half, Sv);

  float sq4[4], qm4[4], sc8[8];
  for (int jt = 0; jt < 4; ++jt) {
    sq4[jt] = sqW[b * kLq + jt * 16 + l16];
    qm4[jt] = qmask[b * kLq + jt * 16 + l16];
  }
  for (int r = 0; r < 8; ++r)
    sc8[r] = scW[b * kLc + ibase + wv * 16 + r + 8 * half];
  for (int jt = 0; jt < 4; ++jt)
    for (int r = 0; r < 8; ++r) Sv[jt][r] += sc8[r] + sq4[jt];

  // Row-wise (axis j) masked softmax stats
  for (int r = 0; r < 8; ++r) {
    float vals[4]; float m = -3.0e38f;
    for (int jt = 0; jt < 4; ++jt) {
      float v = Sv[jt][r] * qm4[jt] + (1.0f - qm4[jt]) * kNegInf;
      vals[jt] = v; m = fmaxf(m, v);
    }
    for (int msk = 1; msk < 16; msk <<= 1) m = fmaxf(m, __shfl_xor(m, msk, 32));
    float s = 0.0f;
    for (int jt = 0; jt < 4; ++jt) s += __expf(vals[jt] - m);
    for (int msk = 1; msk < 16; msk <<= 1) s += __shfl_xor(s, msk, 32);
    if (l16 == 0) {
      int rg = b * kLc + ibase + wv * 16 + r + 8 * half;
      rowmaxW[rg] = m; rowsumW[rg] = s;
    }
  }
  // Column-wise (axis i) partial max on unmasked S
  for (int jt = 0; jt < 4; ++jt) {
    float cm = Sv[jt][0];
    for (int r = 1; r < 8; ++r) cm = fmaxf(cm, Sv[jt][r]);
    cm = fmaxf(cm, __shfl_xor(cm, 16, 32));
    if (half == 0) colL[wv * 64 + jt * 16 + l16] = cm;
  }
  __syncthreads();
  if (tid < 64) {
    float m = colL[tid];
    for (int v2 = 1; v2 < 8; ++v2) m = fmaxf(m, colL[v2 * 64 + tid]);
    atomicMaxF(&colmaxW[b * kLq + tid], m);
  }
}

// ---------------- pass 2: column sums + R = S2num^T @ Ct ----------------

__global__ __launch_bounds__(256)
void k2_colsum_R(const float* __restrict__ C, const float* __restrict__ Q,
                 const float* __restrict__ w,
                 const float* __restrict__ scW, const float* __restrict__ sqW,
                 const float* __restrict__ colmaxW,
                 float* __restrict__ colsumW, float* __restrict__ RW) {
  __shared__ float CtL[128 * 129];
  __shared__ float QwL[64 * 129];
  __shared__ float EL[128 * 65];
  __shared__ float colL[8 * 64];
  const int tid = threadIdx.x;
  const int b = blockIdx.y;
  const int ibase = blockIdx.x * 128;
  const size_t cb = (size_t)b * kD * kLc;
  const size_t qb = (size_t)b * kD * kLq;

  for (int idx = tid; idx < 128 * 128; idx += 256) {
    int i = idx & 127, d = idx >> 7;
    CtL[i * 129 + d] = C[cb + (size_t)d * kLc + ibase + i];
  }
  for (int idx = tid; idx < 64 * 128; idx += 256) {
    int j = idx & 63, d = idx >> 6;
    QwL[j * 129 + d] = Q[qb + (size_t)d * kLq + j] * w[2 * kD + d];
  }
  __syncthreads();

  const int wv = tid >> 5, lane = tid & 31, l16 = lane & 15, half = lane >> 4;
  v8f Sv[4] = {};
  scq_strip<false>(CtL, QwL, nullptr, wv, l16, half, Sv);

  float sq4[4], sc8[8], cm4[4];
  for (int jt = 0; jt < 4; ++jt) {
    sq4[jt] = sqW[b * kLq + jt * 16 + l16];
    cm4[jt] = colmaxW[b * kLq + jt * 16 + l16];
  }
  for (int r = 0; r < 8; ++r)
    sc8[r] = scW[b * kLc + ibase + wv * 16 + r + 8 * half];

  for (int jt = 0; jt < 4; ++jt) {
    float s = 0.0f;
    for (int r = 0; r < 8; ++r) {
      float e = __expf(Sv[jt][r] + sc8[r] + sq4[jt] - cm4[jt]);
      s += e;
      EL[(wv * 16 + r + 8 * half) * 65 + jt * 16 + l16] = e;
    }
    s += __shfl_xor(s, 16, 32);
    if (half == 0) colL[wv * 64 + jt * 16 + l16] = s;
  }
  __syncthreads();
  if (tid < 64) {
    float s = 0.0f;
    for (int v2 = 0; v2 < 8; ++v2) s += colL[v2 * 64 + tid];
    atomicAdd(&colsumW[b * kLq + tid], s);
  }

  // R_partial(64x128) = E^T (64 x 128rows) @ Ct_tile (128rows x 128d), WMMA
  const int mt = wv & 3;            // j tile
  const int ntb = (wv >> 2) * 4;    // d tile base
  v8f Rv[4] = {};
  for (int kb = 0; kb < 32; ++kb) {
    const int k = kb * 4 + half * 2;
    v2f a{};
    a[0] = EL[k * 65 + mt * 16 + l16];
    a[1] = EL[(k + 1) * 65 + mt * 16 + l16];
    for (int q = 0; q < 4; ++q) {
      const int nc = (ntb + q) * 16 + l16;
      v2f bb{};
      bb[0] = CtL[k * 129 + nc];
      bb[1] = CtL[(k + 1) * 129 + nc];
      Rv[q] = wmma4(a, bb, Rv[q]);
    }
  }
  for (int q = 0; q < 4; ++q)
    for (int r = 0; r < 8; ++r) {
      int j = mt * 16 + r + 8 * half;
      int d = (ntb + q) * 16 + l16;
      atomicAdd(&RW[(size_t)b * kLq * kD + j * kD + d], Rv[q][r]);
    }
}

// ---------------- pass 3: S1, A = S1@Qt, Bm = S1@Rn, fused output ----------------

__global__ __launch_bounds__(256)
void k3_output(const float* __restrict__ C, const float* __restrict__ Q,
               const float* __restrict__ w, const float* __restrict__ qmask,
               const float* __restrict__ scW, const float* __restrict__ sqW,
               const float* __restrict__ rowmaxW, const float* __restrict__ rowsumW,
               const float* __restrict__ colsumW, const float* __restrict__ RW,
               float* __restrict__ out) {
  __shared__ float CtL[128 * 129];
  __shared__ float QtL[64 * 129];
  __shared__ float RnL[64 * 129];
  __shared__ float S1L[128 * 65];
  __shared__ float stA[8 * 272];
  __shared__ float stB[8 * 272];
  __shared__ float wl[128];
  const int tid = threadIdx.x;
  const int b = blockIdx.y;
  const int ibase = blockIdx.x * 128;
  const size_t cb = (size_t)b * kD * kLc;
  const size_t qb = (size_t)b * kD * kLq;
  const size_t rb = (size_t)b * kLq * kD;

  for (int idx = tid; idx < 128 * 128; idx += 256) {
    int i = idx & 127, d = idx >> 7;
    CtL[i * 129 + d] = C[cb + (size_t)d * kLc + ibase + i];
  }
  for (int idx = tid; idx < 64 * 128; idx += 256) {
    int j = idx & 63, d = idx >> 6;
    QtL[j * 129 + d] = Q[qb + (size_t)d * kLq + j];
  }
  for (int idx = tid; idx < 64 * 128; idx += 256) {
    int j = idx >> 7, d = idx & 127;
    RnL[j * 129 + d] = RW[rb + j * kD + d] / colsumW[b * kLq + j];
  }
  if (tid < 128) wl[tid] = w[2 * kD + tid];
  __syncthreads();

  const int wv = tid >> 5, lane = tid & 31, l16 = lane & 15, half = lane >> 4;
  v8f Sv[4] = {};
  scq_strip<true>(CtL, QtL, wl, wv, l16, half, Sv);

  float sq4[4], qm4[4], sc8[8], rm8[8], ri8[8];
  for (int jt = 0; jt < 4; ++jt) {
    sq4[jt] = sqW[b * kLq + jt * 16 + l16];
    qm4[jt] = qmask[b * kLq + jt * 16 + l16];
  }
  for (int r = 0; r < 8; ++r) {
    int rg = b * kLc + ibase + wv * 16 + r + 8 * half;
    sc8[r] = scW[rg - b * kLc + b * kLc];  // = scW[rg]
    rm8[r] = rowmaxW[rg];
    ri8[r] = 1.0f / rowsumW[rg];
  }
  for (int jt = 0; jt < 4; ++jt)
    for (int r = 0; r < 8; ++r) {
      float sfull = Sv[jt][r] + sc8[r] + sq4[jt];
      float smask = sfull * qm4[jt] + (1.0f - qm4[jt]) * kNegInf;
      float s1 = __expf(smask - rm8[r]) * ri8[r];
      S1L[(wv * 16 + r + 8 * half) * 65 + jt * 16 + l16] = s1;
    }
  // Each wave consumes only its own 16 S1 rows: per-wave LDS ops are in order.

  v8f accA[8] = {};
  v8f accB[8] = {};
  const int arow = (wv * 16 + l16) * 65;
  for (int kb = 0; kb < 16; ++kb) {       // K = Lq = 64
    const int k = kb * 4 + half * 2;
    v2f a{};
    a[0] = S1L[arow + k];
    a[1] = S1L[arow + k + 1];
    for (int nt = 0; nt < 8; ++nt) {
      const int nc = nt * 16 + l16;
      v2f bq{};
      bq[0] = QtL[k * 129 + nc];
      bq[1] = QtL[(k + 1) * 129 + nc];
      accA[nt] = wmma4(a, bq, accA[nt]);
      v2f br{};
      br[0] = RnL[k * 129 + nc];
      br[1] = RnL[(k + 1) * 129 + nc];
      accB[nt] = wmma4(a, br, accB[nt]);
    }
  }

  // Transpose through per-wave staging tiles; write 4 fused output blocks
  float* sA = stA + wv * 272;
  float* sB = stB + wv * 272;
  const size_t ob = (size_t)b * 4 * kD * kLc;
  for (int nt = 0; nt < 8; ++nt) {
    for (int r = 0; r < 8; ++r) {
      sA[(r + 8 * half) * 17 + l16] = accA[nt][r];
      sB[(r + 8 * half) * 17 + l16] = accB[nt][r];
    }
    for (int it = 0; it < 8; ++it) {
      int idx = it * 32 + lane;
      int i_l = idx & 15, d_l = idx >> 4;
      float va = sA[i_l * 17 + d_l];
      float vb = sB[i_l * 17 + d_l];
      float ct = CtL[(wv * 16 + i_l) * 129 + nt * 16 + d_l];
      int ig = ibase + wv * 16 + i_l;
      int dg = nt * 16 + d_l;
      out[ob + (size_t)dg * kLc + ig]              = ct;
      out[ob + (size_t)(kD + dg) * kLc + ig]       = va;
      out[ob + (size_t)(2 * kD + dg) * kLc + ig]   = ct * va;
      out[ob + (size_t)(3 * kD + dg) * kLc + ig]   = ct * vb;
    }
  }
}

extern "C" void kernel_launch(void* const* d_in, const int* in_sizes, int n_in,
                              void* d_out, int out_size, void* d_ws, size_t ws_size,
                              hipStream_t stream) {
  const float* C     = (const float*)d_in[0];
  const float* Q     = (const float*)d_in[1];
  const float* qmask = (const float*)d_in[2];
  const float* w     = (const float*)d_in[3];
  float* out = (float*)d_out;
  float* ws  = (float*)d_ws;

  float* rowmaxW = ws;                 // B*Lc = 65536
  float* rowsumW = ws + 65536;         // 65536
  float* scW     = ws + 131072;        // 65536
  float* sqW     = ws + 196608;        // 2048
  float* colmaxW = ws + 198656;        // 2048
  float* colsumW = ws + 200704;        // 2048
  float* RW      = ws + 202752;        // B*Lq*D = 262144
  // total ~1.8 MB of workspace

  dim3 blk(256);
  k_init<<<dim3((kB * kLq * kD) / 256), blk, 0, stream>>>(colmaxW, colsumW, RW);
  k_sc<<<dim3((kB * kLc) / 256), blk, 0, stream>>>(C, w, scW);
  k_sq<<<dim3(kB), dim3(64), 0, stream>>>(Q, w, sqW);

  dim3 grid(kLc / 128, kB);
  k1_stats<<<grid, blk, 0, stream>>>(C, Q, w, qmask, scW, sqW, rowmaxW, rowsumW, colmaxW);
  k2_colsum_R<<<grid, blk, 0, stream>>>(C, Q, w, scW, sqW, colmaxW, colsumW, RW);
  k3_output<<<grid, blk, 0, stream>>>(C, Q, w, qmask, scW, sqW, rowmaxW, rowsumW,
                                      colsumW, RW, out);
}